// MoEClassifier_16638703304805
// MI455X (gfx1250) — compile-verified
//
#include <hip/hip_runtime.h>
#include <math.h>

// ---------------------------------------------------------------------------
// MoE classifier, MI455X (gfx1250, wave32, WMMA + TDM async tensor loads).
// B=16384, D=256, E=16, K=4(top-k), C=10.
// Dominant compute: 34.4 GFLOP fp32 -> V_WMMA_F32_16X16X4_F32.
// W1 is pre-scaled by ln_g and pre-pair-packed; staged to LDS by the
// Tensor Data Mover (double-buffered, TENSORcnt-tracked) to overlap with WMMA.
// ---------------------------------------------------------------------------

typedef float v2f __attribute__((ext_vector_type(2)));
typedef float v4f __attribute__((ext_vector_type(4)));
typedef float v8f __attribute__((ext_vector_type(8)));
typedef unsigned int u32x4 __attribute__((ext_vector_type(4)));
typedef int i32x4 __attribute__((ext_vector_type(4)));
typedef int i32x8 __attribute__((ext_vector_type(8)));

#define NB   16384      // batch
#define ND   256        // feature dim
#define NE   16         // experts
#define NC   10         // classes
#define LDA  260        // xhat/h tile stride: 260%64=4 -> conflict-free float2
                        // A-frag loads; 260*4 B rows are 16B aligned.
                        // Produced by TDM pad: 4 dwords after every 256.
#define WBP  544        // pair-row stride: 512 data + 2x16 pad (TDM pads 16
                        // dwords per 256) -> 544%64=32, half-waves disjoint.
#define WBUF (4 * WBP)  // one K-chunk (8 K rows = 4 pair rows) = 2176 floats

// ---- Tensor Data Mover: 2D tile load with per-row LDS padding -------------
// dim0 elems/row (f32), rows, row stride (elems); pad codes per CDNA5 D#.
__device__ __forceinline__ void tdm_load_2d(
    unsigned lds_byte_off, const float* g, unsigned dim0, unsigned rows,
    unsigned stride0, unsigned pad_int_code, unsigned pad_amt_code)
{
    const unsigned long long ga = (unsigned long long)(uintptr_t)g;
    u32x4 g0 = { 1u,                                   // count=1 (valid D#)
                 lds_byte_off,                         // LDS byte address
                 (unsigned)ga,                         // global_addr[31:0]
                 (unsigned)((ga >> 32) & 0x01ffffffu)  // global_addr[56:32]
                     | 0x80000000u };                  // type=2 ("image")
    const int w0 = (2 << 16)                           // data_size = 4B
                 | (1 << 20)                           // pad_enable
                 | ((int)pad_int_code << 22)
                 | ((int)pad_amt_code << 25);
    const int w1 = (int)((dim0 & 0xffffu) << 16);              // tensor_dim0 lo
    const int w2 = (int)((dim0 >> 16) & 0xffffu)
                 | (int)((rows & 0xffffu) << 16);              // tensor_dim1 lo
    const int w3 = (int)((rows >> 16) & 0xffffu)
                 | (int)((dim0 & 0xffffu) << 16);              // tile_dim0
    const int w4 = (int)(rows & 0xffffu);                      // tile_dim1
    const int w5 = (int)stride0;                               // dim0 stride lo
    i32x8 g1 = { w0, w1, w2, w3, w4, w5, 0, 0 };
    i32x4 z4 = { 0, 0, 0, 0 };
    i32x8 z8 = { 0, 0, 0, 0, 0, 0, 0, 0 };
    __builtin_amdgcn_tensor_load_to_lds(g0, g1, z4, z4, z8, 0);
}

// ============================ Kernel 1 =====================================
// One wave per row: LN stats -> xhat, router GEMV -> gate softmax, top-4 ->
// dense mixing weights wmix[B,E] (0 except renormalized top-k).
__global__ __launch_bounds__(256) void k1_router_ln(
    const float* __restrict__ F, const float* __restrict__ RW,
    const float* __restrict__ RB, float* __restrict__ xhat,
    float* __restrict__ gate, float* __restrict__ wmix)
{
    const int wave = threadIdx.x >> 5;
    const int lane = threadIdx.x & 31;
    const long b = (long)blockIdx.x * 8 + wave;

    const float* x = F + b * ND;
    v4f xa = *(const v4f*)(x + lane * 8);
    v4f xb = *(const v4f*)(x + lane * 8 + 4);
    float xv[8] = {xa.x, xa.y, xa.z, xa.w, xb.x, xb.y, xb.z, xb.w};

    float s = 0.f, ss = 0.f;
#pragma unroll
    for (int i = 0; i < 8; ++i) { s += xv[i]; ss += xv[i] * xv[i]; }
#pragma unroll
    for (int m = 16; m >= 1; m >>= 1) {
        s  += __shfl_xor(s,  m, 32);
        ss += __shfl_xor(ss, m, 32);
    }
    const float mu   = s * (1.f / 256.f);
    const float var  = ss * (1.f / 256.f) - mu * mu;
    const float rstd = rsqrtf(var + 1e-5f);

    float xh[8];
#pragma unroll
    for (int i = 0; i < 8; ++i) xh[i] = (xv[i] - mu) * rstd;
    v4f o0 = {xh[0], xh[1], xh[2], xh[3]};
    v4f o1 = {xh[4], xh[5], xh[6], xh[7]};
    *(v4f*)(xhat + b * ND + lane * 8)     = o0;
    *(v4f*)(xhat + b * ND + lane * 8 + 4) = o1;

    // router: g[e] = sum_d x[d] * RW[d][e]
    float g[16];
#pragma unroll
    for (int c = 0; c < 16; ++c) g[c] = 0.f;
#pragma unroll
    for (int i = 0; i < 8; ++i) {
        const float* wr = RW + (size_t)(lane * 8 + i) * 16;
        v4f w0 = *(const v4f*)(wr);
        v4f w1 = *(const v4f*)(wr + 4);
        v4f w2 = *(const v4f*)(wr + 8);
        v4f w3 = *(const v4f*)(wr + 12);
        const float xi = xv[i];
        g[0] += xi * w0.x;  g[1] += xi * w0.y;  g[2]  += xi * w0.z;  g[3]  += xi * w0.w;
        g[4] += xi * w1.x;  g[5] += xi * w1.y;  g[6]  += xi * w1.z;  g[7]  += xi * w1.w;
        g[8] += xi * w2.x;  g[9] += xi * w2.y;  g[10] += xi * w2.z;  g[11] += xi * w2.w;
        g[12]+= xi * w3.x;  g[13]+= xi * w3.y;  g[14] += xi * w3.z;  g[15] += xi * w3.w;
    }
#pragma unroll
    for (int c = 0; c < 16; ++c) {
        float t = g[c];
#pragma unroll
        for (int m = 16; m >= 1; m >>= 1) t += __shfl_xor(t, m, 32);
        g[c] = t + RB[c];
    }

    // softmax over 16 experts
    float mx = g[0];
#pragma unroll
    for (int c = 1; c < 16; ++c) mx = fmaxf(mx, g[c]);
    float den = 0.f;
#pragma unroll
    for (int c = 0; c < 16; ++c) { g[c] = expf(g[c] - mx); den += g[c]; }
    const float inv = 1.f / den;
#pragma unroll
    for (int c = 0; c < 16; ++c) g[c] *= inv;

    if (lane == 0) {
        v4f a0 = {g[0], g[1], g[2], g[3]};
        v4f a1 = {g[4], g[5], g[6], g[7]};
        v4f a2 = {g[8], g[9], g[10], g[11]};
        v4f a3 = {g[12], g[13], g[14], g[15]};
        float* gp = gate + b * 16;
        *(v4f*)(gp)      = a0;  *(v4f*)(gp + 4)  = a1;
        *(v4f*)(gp + 8)  = a2;  *(v4f*)(gp + 12) = a3;
    }

    // top-4 (strict > keeps lowest index on ties, matching lax.top_k)
    float p[16], w[16];
#pragma unroll
    for (int c = 0; c < 16; ++c) { p[c] = g[c]; w[c] = 0.f; }
    float tsum = 0.f;
#pragma unroll
    for (int k = 0; k < 4; ++k) {
        float mv = -1.f; int mi = 0;
#pragma unroll
        for (int c = 0; c < 16; ++c) if (p[c] > mv) { mv = p[c]; mi = c; }
        tsum += mv;
#pragma unroll
        for (int c = 0; c < 16; ++c) if (c == mi) { w[c] = mv; p[c] = -2.f; }
    }
    const float winv = 1.f / tsum;
    if (lane == 0) {
        v4f b0 = {w[0]*winv,  w[1]*winv,  w[2]*winv,  w[3]*winv};
        v4f b1 = {w[4]*winv,  w[5]*winv,  w[6]*winv,  w[7]*winv};
        v4f b2 = {w[8]*winv,  w[9]*winv,  w[10]*winv, w[11]*winv};
        v4f b3 = {w[12]*winv, w[13]*winv, w[14]*winv, w[15]*winv};
        float* wp = wmix + b * 16;
        *(v4f*)(wp)      = b0;  *(v4f*)(wp + 4)  = b1;
        *(v4f*)(wp + 8)  = b2;  *(v4f*)(wp + 12) = b3;
    }
}

// ============================ Kernel 2a ====================================
// W1pp: ln_g-scaled AND pair-packed: element (e,k,n) ->
//   W1pp[e*65536 + (k>>1)*512 + 2n + (k&1)]
// so every 8-K chunk is one contiguous 8KB block (TDM-friendly).
__global__ __launch_bounds__(256) void k2a_pack_w1(
    const float* __restrict__ w1, const float* __restrict__ ln_g,
    float* __restrict__ W1pp)
{
    const long f  = (long)blockIdx.x * 256 + threadIdx.x;  // float4 out index
    const int  x4 = (int)(f & 127);        // 128 float4 per pair-row
    const long pg = f >> 7;                // e*128 + pair_row
    const long r0 = pg * 2;                // even k row (= e*256 + 2*pr)
    v2f a0 = *(const v2f*)(w1 + r0 * 256 + 2 * x4);
    v2f a1 = *(const v2f*)(w1 + (r0 + 1) * 256 + 2 * x4);
    const float s0 = ln_g[r0], s1 = ln_g[r0 + 1];
    v4f o = { a0.x * s0, a1.x * s1, a0.y * s0, a1.y * s1 };
    ((v4f*)W1pp)[f] = o;
}

// ============================ Kernel 2b ====================================
// b1p[e,h] = b1[e,h] + sum_d ln_b[e,d] * w1[e,d,h]
__global__ __launch_bounds__(256) void k2b_fold_b1(
    const float* __restrict__ w1, const float* __restrict__ ln_b,
    const float* __restrict__ b1, float* __restrict__ b1p)
{
    const int e = blockIdx.x, h = threadIdx.x;
    const float* W  = w1 + (size_t)e * ND * ND;
    const float* lb = ln_b + e * ND;
    float s = b1[e * ND + h];
    for (int d = 0; d < ND; ++d) s += lb[d] * W[(size_t)d * ND + h];
    b1p[e * ND + h] = s;
}

// ============================ Kernel 2c ====================================
// Pad w2 [E,256,10] to 16 cols and pair-pack K:
// W2pp[e*4096 + (k>>1)*32 + 2c + (k&1)] = (c<10) ? w2[e,k,c] : 0
__global__ __launch_bounds__(256) void k2c_pack_w2(
    const float* __restrict__ w2, float* __restrict__ W2pp)
{
    const int o = blockIdx.x * 256 + threadIdx.x;   // 0..65535
    const int c = o & 15, g = o >> 4;               // g = e*256 + k
    const int e = g >> 8, k = g & 255;
    const float v = (c < NC) ? w2[(size_t)g * NC + c] : 0.f;
    W2pp[e * 4096 + (k >> 1) * 32 + 2 * c + (k & 1)] = v;
}

// ============================ Kernel 3 =====================================
// Per (32-row tile, expert): h = gelu(xhat@W1 + b1'); logits = h@W2 + b2;
// all_probs = softmax_C(logits). xhat tile + W1 chunks staged by TDM.
__global__ __launch_bounds__(256) void k3_moe_main(
    const float* __restrict__ xhat, const float* __restrict__ W1pp,
    const float* __restrict__ b1p, const float* __restrict__ W2pp,
    const float* __restrict__ b2, float* __restrict__ all_probs)
{
    __shared__ float XA[32 * LDA];   // xhat tile; reused as h tile after GEMM1
    __shared__ float WB[2 * WBUF];   // double-buffered W1 chunk; RED in GEMM2

    const int tid  = threadIdx.x;
    const int wave = tid >> 5;
    const int lane = tid & 31;
    const int lm   = lane & 15;      // N-lane / M-lane within tile
    const int hi   = lane >> 4;      // half-wave selector (K/M offset)
    const int e    = blockIdx.y;
    const int row0 = blockIdx.x * 32;

    const float* W1e = W1pp + (size_t)e * ND * ND;
    const unsigned xa_lds = (unsigned)(uintptr_t)(&XA[0]);
    const unsigned wb_lds = (unsigned)(uintptr_t)(&WB[0]);

    // Prologue: async-stage xhat tile (2D, pad 4/row -> stride 260) and
    // chunk 0 of W1 (1D contiguous 2048 f32, pad 16/256 -> pair stride 544).
    if (wave == 0) {
        tdm_load_2d(xa_lds, xhat + (size_t)row0 * ND, ND, 32, ND, 7, 3);
        tdm_load_2d(wb_lds, W1e, 2048, 1, 2048, 7, 15);
    }

    const int mt = wave & 1;   // M tile (16 rows)
    const int nq = wave >> 1;  // N quarter (64 cols)
    const v8f vz = {0.f, 0.f, 0.f, 0.f, 0.f, 0.f, 0.f, 0.f};
    v8f acc[4] = {vz, vz, vz, vz};

    for (int c = 0; c < 32; ++c) {          // 32 chunks x 8 K
        if (wave == 0) __builtin_amdgcn_s_wait_tensorcnt((short)0);
        __syncthreads();                    // chunk c (and XA) ready for all
        if (wave == 0 && c + 1 < 32)        // prefetch next chunk -> alt buf
            tdm_load_2d(wb_lds + 4u * (unsigned)(((c + 1) & 1) * WBUF),
                        W1e + (size_t)(c + 1) * 2048, 2048, 1, 2048, 7, 15);
        const float* WBc = WB + (c & 1) * WBUF;
#pragma unroll
        for (int kk = 0; kk < 2; ++kk) {
            v2f a = *(const v2f*)(&XA[(mt * 16 + lm) * LDA + c * 8 + kk * 4 + 2 * hi]);
#pragma unroll
            for (int nt = 0; nt < 4; ++nt) {
                const int n = nq * 64 + nt * 16 + lm;
                // pair-packed + mid-row TDM pad: +16 dwords once n >= 128
                const int off = (kk * 2 + hi) * WBP + 2 * n + ((n >> 7) << 4);
                v2f bv = *(const v2f*)(WBc + off);
                acc[nt] = __builtin_amdgcn_wmma_f32_16x16x4_f32(
                    false, a, false, bv, (short)0, acc[nt], false, false);
            }
        }
        __syncthreads();                    // done reading buf before recycle
    }

    // ---- bias + exact-erf GELU, store h into XA (alias) ----
#pragma unroll
    for (int nt = 0; nt < 4; ++nt) {
        const int n  = nq * 64 + nt * 16 + lm;
        const float bb = b1p[e * ND + n];
#pragma unroll
        for (int r = 0; r < 8; ++r) {
            const float pre = acc[nt][r] + bb;
            const float hv  = 0.5f * pre * (1.f + erff(pre * 0.70710678118654752f));
            XA[(mt * 16 + r + hi * 8) * LDA + n] = hv;
        }
    }
    __syncthreads();

    // ---- GEMM2: logits[32x16] = h @ W2pp, split-K over 4 wave-groups ----
    const int mt2 = wave & 1;
    const int kq  = wave >> 1;       // K quarter (64)
    v8f acc2 = vz;
    const float* W2e = W2pp + e * 4096;
#pragma unroll
    for (int kk = 0; kk < 16; ++kk) {
        const int k = kq * 64 + kk * 4;
        v2f a  = *(const v2f*)(&XA[(mt2 * 16 + lm) * LDA + k + 2 * hi]);
        v2f bv = *(const v2f*)(W2e + ((k >> 1) + hi) * 32 + 2 * lm);
        acc2 = __builtin_amdgcn_wmma_f32_16x16x4_f32(
            false, a, false, bv, (short)0, acc2, false, false);
    }
    __syncthreads();   // WB safe to reuse as reduction buffer
    if (kq != 0) {
        const int base = (wave - 2) * 8 * 32;
#pragma unroll
        for (int r = 0; r < 8; ++r) WB[base + r * 32 + lane] = acc2[r];
    }
    __syncthreads();
    if (kq == 0) {
#pragma unroll
        for (int j = 1; j < 4; ++j) {
            const int pw   = 2 * j + mt2;
            const int base = (pw - 2) * 8 * 32;
#pragma unroll
            for (int r = 0; r < 8; ++r) acc2[r] += WB[base + r * 32 + lane];
        }
        const float bias = (lm < NC) ? b2[e * NC + lm] : 0.f;
#pragma unroll
        for (int r = 0; r < 8; ++r) {
            float v = acc2[r] + bias;
            v = (lm < NC) ? v : -3.0e38f;
            float m = v;
#pragma unroll
            for (int s2 = 8; s2 >= 1; s2 >>= 1) m = fmaxf(m, __shfl_xor(m, s2, 16));
            float p = (lm < NC) ? expf(v - m) : 0.f;
            float sum = p;
#pragma unroll
            for (int s2 = 8; s2 >= 1; s2 >>= 1) sum += __shfl_xor(sum, s2, 16);
            const float prob = p / sum;
            const int brow = row0 + mt2 * 16 + r + hi * 8;
            if (lm < NC)
                all_probs[((size_t)e * NB + brow) * NC + lm] = prob;
        }
    }
}

// ============================ Kernel 4 =====================================
// market[b,c] = sum_e wmix[b,e] * all_probs[e,b,c]
__global__ __launch_bounds__(256) void k4_mix(
    const float* __restrict__ allp, const float* __restrict__ wmix,
    float* __restrict__ market)
{
    const int t = blockIdx.x * 256 + threadIdx.x;
    if (t >= NB * NC) return;
    const int b = t / NC, c = t - b * NC;
    float s = 0.f;
#pragma unroll
    for (int e = 0; e < NE; ++e)
        s += wmix[b * NE + e] * allp[((size_t)e * NB + b) * NC + c];
    market[t] = s;
}

// ============================ Launch =======================================
extern "C" void kernel_launch(void* const* d_in, const int* in_sizes, int n_in,
                              void* d_out, int out_size, void* d_ws, size_t ws_size,
                              hipStream_t stream) {
    (void)in_sizes; (void)n_in; (void)out_size; (void)ws_size;

    const float* features = (const float*)d_in[0];   // [B,D]
    const float* router_w = (const float*)d_in[1];   // [D,E]
    const float* router_b = (const float*)d_in[2];   // [E]
    const float* ln_g     = (const float*)d_in[3];   // [E,D]
    const float* ln_b     = (const float*)d_in[4];   // [E,D]
    const float* w1       = (const float*)d_in[5];   // [E,D,D]
    const float* b1       = (const float*)d_in[6];   // [E,D]
    const float* w2       = (const float*)d_in[7];   // [E,D,C]
    const float* b2       = (const float*)d_in[8];   // [E,C]

    float* out    = (float*)d_out;
    float* market = out;                              // [B,C]
    float* allp   = out + (size_t)NB * NC;            // [E,B,C]
    float* gate   = allp + (size_t)NE * NB * NC;      // [B,E]

    float* ws   = (float*)d_ws;
    float* xhat = ws;                                 // B*D
    float* W1pp = xhat + (size_t)NB * ND;             // E*D*D (pair-packed)
    float* b1p  = W1pp + (size_t)NE * ND * ND;        // E*D
    float* W2pp = b1p + (size_t)NE * ND;              // E*256*16
    float* wmix = W2pp + (size_t)NE * ND * 16;        // B*E

    k1_router_ln<<<NB / 8, 256, 0, stream>>>(features, router_w, router_b,
                                             xhat, gate, wmix);
    k2a_pack_w1<<<(NE * ND * ND / 4) / 256, 256, 0, stream>>>(w1, ln_g, W1pp);
    k2b_fold_b1<<<NE, 256, 0, stream>>>(w1, ln_b, b1, b1p);
    k2c_pack_w2<<<(NE * ND * 16) / 256, 256, 0, stream>>>(w2, W2pp);
    k3_moe_main<<<dim3(NB / 32, NE), 256, 0, stream>>>(xhat, W1pp, b1p, W2pp,
                                                       b2, allp);
    k4_mix<<<(NB * NC + 255) / 256, 256, 0, stream>>>(allp, wmix, market);
}